// GCN_39960375722198
// MI455X (gfx1250) — compile-verified
//
#include <hip/hip_runtime.h>

typedef __attribute__((ext_vector_type(2))) float v2f;
typedef __attribute__((ext_vector_type(8))) float v8f;

#define N_NODES_C 100000
#define K_DIM 256
#define TB 256            // threads per block (8 waves)
#define ROWS_PER_BLOCK 128
#define KCHUNK 64
#define KPAD 68           // LDS row stride in floats: (4r + k) % 64 distinct -> no bank conflicts

// ---------------- K1: degree init (self loop contributes 1) ----------------
__global__ void k_deg_init(float* __restrict__ deg, int n) {
  int i = blockIdx.x * blockDim.x + threadIdx.x;
  if (i < n) deg[i] = 1.0f;
}

// ---------------- K2: in-degree count over edges ----------------
__global__ void k_deg_count(const long long* __restrict__ dst, float* __restrict__ deg, int e) {
  int i = blockIdx.x * blockDim.x + threadIdx.x;
  if (i < e) atomicAdd(&deg[(int)dst[i]], 1.0f);
}

// ---------------- K3: dinv = rsqrt(deg), in place ----------------
__global__ void k_dinv(float* __restrict__ deg, int n) {
  int i = blockIdx.x * blockDim.x + threadIdx.x;
  if (i < n) deg[i] = rsqrtf(deg[i]);
}

// ---------------- K4: h_pre = x @ W1 via V_WMMA_F32_16X16X4_F32 ------------
// Each wave owns one 16-row tile (N is a multiple of 16 -> tiles never straddle).
// Epilogue also seeds agg with the self-loop term h_pre * dinv^2.
__global__ void __launch_bounds__(TB)
k_gemm_wmma(const float* __restrict__ x, const float* __restrict__ W1,
            const float* __restrict__ dinv, float* __restrict__ hpre,
            float* __restrict__ agg, int n) {
  __shared__ float xs[ROWS_PER_BLOCK * KPAD];   // ~34 KB staged x tile
  __shared__ v2f   bs[(K_DIM / 4) * 32];        // 16 KB: B fragments, W1 zero-padded to 16 cols

  const int t      = threadIdx.x;
  const int lane   = t & 31;
  const int wave   = t >> 5;
  const int laneN  = lane & 15;
  const int hiHalf = (lane >> 4) & 1;

  // Build B fragments once. B is 4x16 (KxN): VGPR0 holds K=(step*4 + 2*hi), VGPR1 next K.
  for (int i = t; i < (K_DIM / 4) * 32; i += TB) {
    int step = i >> 5;
    int l    = i & 31;
    int nn   = l & 15;
    int k    = step * 4 + ((l & 16) ? 2 : 0);
    v2f b;
    b.x = (nn < 3) ? W1[k * 3 + nn]       : 0.0f;
    b.y = (nn < 3) ? W1[(k + 1) * 3 + nn] : 0.0f;
    bs[i] = b;
  }

  const int blockRow = blockIdx.x * ROWS_PER_BLOCK;
  const int tileRow  = blockRow + wave * 16;
  const bool active  = (tileRow < n);

  v8f acc = {0.f, 0.f, 0.f, 0.f, 0.f, 0.f, 0.f, 0.f};

  for (int kc = 0; kc < K_DIM / KCHUNK; ++kc) {
    // Cooperative coalesced load of 128x64 fp32 chunk (float4 global, float2 LDS stores).
    #pragma unroll
    for (int j = 0; j < (ROWS_PER_BLOCK * KCHUNK / 4) / TB; ++j) {  // 8 iters
      int idx  = t + j * TB;        // 0..2047
      int r    = idx >> 4;          // row within block (16 float4 per row)
      int c4   = (idx & 15) << 2;   // column 0..60
      int grow = blockRow + r;
      float4 v = make_float4(0.f, 0.f, 0.f, 0.f);
      if (grow < n) {
        const float* gp = x + (size_t)grow * K_DIM + kc * KCHUNK + c4;
        v = *(const float4*)gp;
        if (kc + 1 < K_DIM / KCHUNK)
          __builtin_prefetch((const void*)(gp + KCHUNK), 0, 0);  // global_prefetch_b8
      }
      float* lp = &xs[r * KPAD + c4];
      *(float2*)(lp + 0) = make_float2(v.x, v.y);
      *(float2*)(lp + 2) = make_float2(v.z, v.w);
    }
    __syncthreads();

    if (active) {  // wave-uniform branch: EXEC stays all-ones for WMMA
      const int rowOff = (wave * 16 + laneN) * KPAD;
      #pragma unroll
      for (int s = 0; s < KCHUNK / 4; ++s) {
        int kA = s * 4 + hiHalf * 2;
        v2f a;
        a.x = xs[rowOff + kA];
        a.y = xs[rowOff + kA + 1];
        v2f b = bs[(kc * (KCHUNK / 4) + s) * 32 + lane];
        acc = __builtin_amdgcn_wmma_f32_16x16x4_f32(
            /*neg_a=*/false, a, /*neg_b=*/false, b,
            /*c_mod=*/(short)0, acc, /*reuse_a=*/false, /*reuse_b=*/false);
      }
    }
    __syncthreads();
  }

  // C/D layout: VGPR i -> M = i + 8*hiHalf, N = laneN. Only N<3 is real data.
  if (active && laneN < 3) {
    #pragma unroll
    for (int i = 0; i < 8; ++i) {
      int row  = tileRow + hiHalf * 8 + i;
      float v  = acc[i];
      float di = dinv[row];
      hpre[row * 3 + laneN] = v;
      agg[row * 3 + laneN]  = v * di * di;  // self-loop term + agg initialization
    }
  }
}

// ---------------- K5: edge aggregation (L2-resident f32 atomics) -----------
__global__ void k_aggregate(const long long* __restrict__ src,
                            const long long* __restrict__ dst,
                            const float* __restrict__ dinv,
                            const float* __restrict__ hpre,
                            float* __restrict__ agg, int e) {
  int i = blockIdx.x * blockDim.x + threadIdx.x;
  if (i >= e) return;
  int s = (int)src[i];
  int d = (int)dst[i];
  float norm = dinv[s] * dinv[d];
  float h0 = hpre[s * 3 + 0], h1 = hpre[s * 3 + 1], h2 = hpre[s * 3 + 2];
  atomicAdd(&agg[d * 3 + 0], h0 * norm);
  atomicAdd(&agg[d * 3 + 1], h1 * norm);
  atomicAdd(&agg[d * 3 + 2], h2 * norm);
}

// ---------------- K6: h = relu(agg + b1); z = h @ W2 + b2 ------------------
__global__ void k_finalize(const float* __restrict__ agg,
                           const float* __restrict__ b1,
                           const float* __restrict__ W2,
                           const float* __restrict__ b2,
                           float* __restrict__ out_h,
                           float* __restrict__ out_z, int n) {
  int i = blockIdx.x * blockDim.x + threadIdx.x;
  if (i >= n) return;
  float h0 = fmaxf(agg[i * 3 + 0] + b1[0], 0.f);
  float h1 = fmaxf(agg[i * 3 + 1] + b1[1], 0.f);
  float h2 = fmaxf(agg[i * 3 + 2] + b1[2], 0.f);
  out_h[i * 3 + 0] = h0;
  out_h[i * 3 + 1] = h1;
  out_h[i * 3 + 2] = h2;
  #pragma unroll
  for (int j = 0; j < 7; ++j) {
    float z = b2[j] + h0 * W2[j] + h1 * W2[7 + j] + h2 * W2[14 + j];
    out_z[i * 7 + j] = z;
  }
}

extern "C" void kernel_launch(void* const* d_in, const int* in_sizes, int n_in,
                              void* d_out, int out_size, void* d_ws, size_t ws_size,
                              hipStream_t stream) {
  const float*     x    = (const float*)d_in[0];
  const long long* edge = (const long long*)d_in[1];
  const float*     W1   = (const float*)d_in[2];
  const float*     b1   = (const float*)d_in[3];
  const float*     W2   = (const float*)d_in[4];
  const float*     b2   = (const float*)d_in[5];

  const int n = in_sizes[0] / K_DIM;   // 100000
  const int e = in_sizes[1] / 2;       // 3200000
  const long long* src = edge;
  const long long* dst = edge + e;

  // Workspace layout: deg/dinv [n] | hpre [3n] | agg [3n]  (2.8 MB, L2-resident)
  float* ws       = (float*)d_ws;
  float* deg_dinv = ws;
  float* hpre     = ws + n;
  float* agg      = ws + 4 * (size_t)n;

  float* out_h = (float*)d_out;
  float* out_z = out_h + 3 * (size_t)n;

  k_deg_init<<<(n + 255) / 256, 256, 0, stream>>>(deg_dinv, n);
  k_deg_count<<<(e + 255) / 256, 256, 0, stream>>>(dst, deg_dinv, e);
  k_dinv<<<(n + 255) / 256, 256, 0, stream>>>(deg_dinv, n);

  const int tiles  = (n + 15) / 16;          // 6250 wave tiles
  const int blocks = (tiles + 7) / 8;        // 8 waves per block
  k_gemm_wmma<<<blocks, TB, 0, stream>>>(x, W1, deg_dinv, hpre, agg, n);

  k_aggregate<<<(e + 255) / 256, 256, 0, stream>>>(src, dst, deg_dinv, hpre, agg, e);
  k_finalize<<<(n + 255) / 256, 256, 0, stream>>>(agg, b1, W2, b2, out_h, out_z, n);
}